// EdgeMapping_52441550684608
// MI455X (gfx1250) — compile-verified
//
#include <hip/hip_runtime.h>
#include <hip/hip_bf16.h>

typedef float v2f __attribute__((ext_vector_type(2)));
typedef float v8f __attribute__((ext_vector_type(8)));

#define B_  8
#define N_  512
#define D_  128
#define M_  ((N_ * (N_ - 1)) / 2)        // 130816 edges per batch
#define NT_ (N_ / 16)                    // 32 tile rows/cols
#define TILES_ ((NT_ * (NT_ + 1)) / 2)   // 528 upper-triangular 16x16 tiles

// One wave32 per 16x16 tile of G = E * E^T (upper-triangular tiles only).
// Uses V_WMMA_F32_16X16X4_F32: A = rows[i-block] (16x4), B = rows[j-block]^T (4x16).
// For a Gram matrix the A and B fragments have identical per-lane load patterns:
//   frag = float2 at E[base + (lane&15)][k0 + 2*(lane>>4)]
__global__ __launch_bounds__(256) void gram_triu_wmma(
    const float* __restrict__ emb,     // [B, N, D] fp32
    float* __restrict__ logits,        // [B, M]    fp32
    int* __restrict__ combos)          // [M, 2]    int32
{
    const int wid  = blockIdx.x * (blockDim.x >> 5) + (threadIdx.x >> 5);
    const int lane = threadIdx.x & 31;
    const int b    = wid / TILES_;
    if (b >= B_) return;                 // wave-uniform guard (never splits EXEC)
    int rem = wid % TILES_;

    // decode upper-triangular tile coordinates (ti <= tj); wave-uniform
    int ti = 0;
    while (rem >= NT_ - ti) { rem -= NT_ - ti; ++ti; }
    const int tj = ti + rem;

    const int l15 = lane & 15;
    const int h   = lane >> 4;
    const int rowBase = ti * 16;
    const int colBase = tj * 16;

    // per-lane base pointers; K offset 2*h selects the lane-half's K pair
    const float* __restrict__ arow = emb + ((size_t)b * N_ + rowBase + l15) * D_ + 2 * h;
    const float* __restrict__ brow = emb + ((size_t)b * N_ + colBase + l15) * D_ + 2 * h;

    v8f c = {0.f, 0.f, 0.f, 0.f, 0.f, 0.f, 0.f, 0.f};
#pragma unroll
    for (int k0 = 0; k0 < D_; k0 += 4) {
        v2f a  = *(const v2f*)(arow + k0);   // global_load_b64
        v2f bb = *(const v2f*)(brow + k0);   // global_load_b64
        // 8 args: (neg_a, A, neg_b, B, c_mod, C, reuse_a, reuse_b)
        c = __builtin_amdgcn_wmma_f32_16x16x4_f32(
                false, a, false, bb, (short)0, c, false, false);
    }

    // C/D layout: VGPR r -> (M = r + 8*h, N = l15)
#pragma unroll
    for (int r = 0; r < 8; ++r) {
        const int i = rowBase + r + 8 * h;
        const int j = colBase + l15;
        if (j > i) {
            // packed strict-upper-triangle index (row-major over i)
            const int e = i * N_ - ((i * (i + 1)) >> 1) + (j - i - 1);
            logits[(size_t)b * M_ + e] = c[r];
            if (b == 0) {            // emit node_combinations once
                combos[2 * e]     = i;
                combos[2 * e + 1] = j;
            }
        }
    }
}

extern "C" void kernel_launch(void* const* d_in, const int* in_sizes, int n_in,
                              void* d_out, int out_size, void* d_ws, size_t ws_size,
                              hipStream_t stream) {
    const float* emb = (const float*)d_in[0];
    float* logits = (float*)d_out;
    // node_combinations (int32) immediately follow the B*M fp32 logits
    int* combos = (int*)(logits + (size_t)B_ * M_);

    const int totalWaves    = B_ * TILES_;   // 4224
    const int wavesPerBlock = 8;             // 256 threads, 8 wave32s per block
    dim3 grid((totalWaves + wavesPerBlock - 1) / wavesPerBlock);
    dim3 block(wavesPerBlock * 32);

    gram_triu_wmma<<<grid, block, 0, stream>>>(emb, logits, combos);
}